// AngleEnsemble_66116726554739
// MI455X (gfx1250) — compile-verified
//
#include <hip/hip_runtime.h>
#include <hip/hip_bf16.h>
#include <math.h>

#define N_BINS   360
#define M_MIX    3
#define KAPPA_F  10.0f
#define MS_ITERS 10

typedef float v2f __attribute__((ext_vector_type(2)));
typedef float v8f __attribute__((ext_vector_type(8)));

__device__ __forceinline__ float wave_sum32(float v) {
#pragma unroll
    for (int m = 16; m >= 1; m >>= 1) v += __shfl_xor(v, m, 32);
    return v;
}
__device__ __forceinline__ float wave_max32(float v) {
#pragma unroll
    for (int m = 16; m >= 1; m >>= 1) v = fmaxf(v, __shfl_xor(v, m, 32));
    return v;
}

// ---------------------------------------------------------------------------
// Kernel 1: softmax mixture + von Mises mean-shift. One wave32 per batch row.
// Bin mapping is blocked for b128 loads: group g=0..2, elem i=0..3,
//   n = 4*(lane + 32*g) + i   (covers 0..383; n>=360 padded with -inf).
// Each lane issues 3 global_load_b128 per mixture row (512 B per wave-load)
// instead of 12 b32 loads -> 4x fewer VMEM issues on the 283 MB stream.
// ---------------------------------------------------------------------------
__global__ void __launch_bounds__(256)
ms_kernel(const float* __restrict__ von, const float* __restrict__ sin_vec,
          float* __restrict__ fused, int B) {
    const int wave = blockIdx.x * 8 + (threadIdx.x >> 5);
    const int lane = threadIdx.x & 31;
    if (wave >= B) return;
    const int b = wave;
    const float NEG_INF = -__builtin_inff();

    // per-lane bin tables: sin/cos of 2*pi*n/360
    float sn[3][4], cn[3][4];
#pragma unroll
    for (int g = 0; g < 3; ++g) {
#pragma unroll
        for (int i = 0; i < 4; ++i) {
            const int n = 4 * (lane + 32 * g) + i;
            const float t = 6.28318530717958647692f * (float)n / (float)N_BINS;
            __sincosf(t, &sn[g][i], &cn[g][i]);
        }
    }

    // accumulate ws = w*sin_n, wc = w*cos_n over the M=3 softmaxed rows
    float ws[3][4], wc[3][4];
#pragma unroll
    for (int g = 0; g < 3; ++g)
#pragma unroll
        for (int i = 0; i < 4; ++i) { ws[g][i] = 0.0f; wc[g][i] = 0.0f; }

    const float invM = 1.0f / (float)M_MIX;
#pragma unroll 1
    for (int m = 0; m < M_MIX; ++m) {
        const float* row = von + ((size_t)m * (size_t)B + (size_t)b) * N_BINS;
        float x[3][4];
        float mx = NEG_INF;
#pragma unroll
        for (int g = 0; g < 3; ++g) {
            const int base = 4 * (lane + 32 * g);
            if (base < N_BINS) {           // all-or-nothing per float4 group
                const float4 t4 = *reinterpret_cast<const float4*>(row + base);
                x[g][0] = t4.x; x[g][1] = t4.y; x[g][2] = t4.z; x[g][3] = t4.w;
            } else {
                x[g][0] = NEG_INF; x[g][1] = NEG_INF;
                x[g][2] = NEG_INF; x[g][3] = NEG_INF;
            }
#pragma unroll
            for (int i = 0; i < 4; ++i) mx = fmaxf(mx, x[g][i]);
        }
        mx = wave_max32(mx);
        float s = 0.0f;
#pragma unroll
        for (int g = 0; g < 3; ++g)
#pragma unroll
            for (int i = 0; i < 4; ++i) {
                x[g][i] = __expf(x[g][i] - mx);   // exp(-inf)=0 pads invalid bins
                s += x[g][i];
            }
        s = wave_sum32(s);
        const float scale = invM / s;
#pragma unroll
        for (int g = 0; g < 3; ++g)
#pragma unroll
            for (int i = 0; i < 4; ++i) {
                const float w = x[g][i] * scale;
                ws[g][i] = fmaf(w, sn[g][i], ws[g][i]);
                wc[g][i] = fmaf(w, cn[g][i], wc[g][i]);
            }
    }

    // init theta at weighted circular mean
    float S = 0.0f, C = 0.0f;
#pragma unroll
    for (int g = 0; g < 3; ++g)
#pragma unroll
        for (int i = 0; i < 4; ++i) { S += ws[g][i]; C += wc[g][i]; }
    S = wave_sum32(S);
    C = wave_sum32(C);
    float theta = atan2f(S, C);

    // 10 mean-shift iterations: exp(kappa*cos(theta - theta_n)) via fma
#pragma unroll 1
    for (int it = 0; it < MS_ITERS; ++it) {
        float st, ct;
        __sincosf(theta, &st, &ct);
        const float ka = KAPPA_F * ct, kb = KAPPA_F * st;
        float S2 = 0.0f, C2 = 0.0f;
#pragma unroll
        for (int g = 0; g < 3; ++g)
#pragma unroll
            for (int i = 0; i < 4; ++i) {
                const float k = __expf(fmaf(ka, cn[g][i], kb * sn[g][i]));
                S2 = fmaf(k, ws[g][i], S2);
                C2 = fmaf(k, wc[g][i], C2);
            }
        S2 = wave_sum32(S2);
        C2 = wave_sum32(C2);
        theta = atan2f(S2, C2);   // identical on all lanes (butterfly)
    }

    if (lane == 0) {
        float sa, ca;
        __sincosf(theta, &sa, &ca);
        float4 f;
        f.x = sin_vec[2 * b + 0];
        f.y = sin_vec[2 * b + 1];
        f.z = ca;
        f.w = sa;
        reinterpret_cast<float4*>(fused)[b] = f;
    }
}

// ---------------------------------------------------------------------------
// Kernel 2: fused[16x4] @ W1[4x128] via 8x V_WMMA_F32_16X16X4_F32 per tile,
// bias+ReLU on the D fragment, then h @ W2[128x2] folded as per-lane FMAs
// (each lane owns one h column) + 16-lane butterfly reduction, normalize.
// One wave32 per 16-row tile.
// ---------------------------------------------------------------------------
__global__ void __launch_bounds__(256)
mlp_wmma_kernel(const float* __restrict__ fused, const float* __restrict__ W1,
                const float* __restrict__ b1, const float* __restrict__ W2,
                const float* __restrict__ b2, float* __restrict__ out,
                int ntiles) {
    const int wave = blockIdx.x * 8 + (threadIdx.x >> 5);
    const int lane = threadIdx.x & 31;
    if (wave >= ntiles) return;
    const int row0 = wave * 16;
    const int half = lane >> 4;   // 0: rows/K low half, 1: high half
    const int l16  = lane & 15;

    // A fragment (16x4 f32): lanes 0-15 hold row=l16, K={0,1}; lanes 16-31 K={2,3}
    const int k0 = half * 2;
    v2f afrag;
    afrag.x = fused[(row0 + l16) * 4 + k0 + 0];
    afrag.y = fused[(row0 + l16) * 4 + k0 + 1];

    float acc0[8], acc1[8];
#pragma unroll
    for (int r = 0; r < 8; ++r) { acc0[r] = 0.0f; acc1[r] = 0.0f; }

#pragma unroll
    for (int c = 0; c < 8; ++c) {
        const int col = c * 16 + l16;
        // B fragment (4x16 f32): mirrors A's K split across lane halves
        v2f bfrag;
        bfrag.x = W1[(k0 + 0) * 128 + col];
        bfrag.y = W1[(k0 + 1) * 128 + col];
        v8f cz = {};
        // D = A x B + 0  ->  h tile columns [16c .. 16c+15]
        v8f d = __builtin_amdgcn_wmma_f32_16x16x4_f32(
            false, afrag, false, bfrag, (short)0, cz, false, false);

        const float bias = b1[col];
        const float w20  = W2[col * 2 + 0];
        const float w21  = W2[col * 2 + 1];
#pragma unroll
        for (int r = 0; r < 8; ++r) {
            // D layout: this lane holds N=col, M = half*8 + r
            float h = fmaxf(d[r] + bias, 0.0f);      // bias + ReLU
            acc0[r] = fmaf(h, w20, acc0[r]);
            acc1[r] = fmaf(h, w21, acc1[r]);
        }
    }

    // reduce across the 16 lanes of each half (xor masks < 16 stay in-half)
#pragma unroll
    for (int m = 8; m >= 1; m >>= 1) {
#pragma unroll
        for (int r = 0; r < 8; ++r) {
            acc0[r] += __shfl_xor(acc0[r], m, 32);
            acc1[r] += __shfl_xor(acc1[r], m, 32);
        }
    }

    if (l16 == 0) {   // lane 0 -> rows 0-7, lane 16 -> rows 8-15
        const float bx = b2[0], by = b2[1];
#pragma unroll
        for (int r = 0; r < 8; ++r) {
            const int row = row0 + half * 8 + r;
            const float o0 = acc0[r] + bx;
            const float o1 = acc1[r] + by;
            const float nrm = fmaxf(sqrtf(o0 * o0 + o1 * o1), 1e-12f);
            out[row * 2 + 0] = o0 / nrm;
            out[row * 2 + 1] = o1 / nrm;
        }
    }
}

extern "C" void kernel_launch(void* const* d_in, const int* in_sizes, int n_in,
                              void* d_out, int out_size, void* d_ws, size_t ws_size,
                              hipStream_t stream) {
    const float* von  = (const float*)d_in[0];  // [3, B, 360]
    const float* sinv = (const float*)d_in[1];  // [B, 2]
    const float* W1   = (const float*)d_in[2];  // [4, 128]
    const float* b1   = (const float*)d_in[3];  // [128]
    const float* W2   = (const float*)d_in[4];  // [128, 2]
    const float* b2   = (const float*)d_in[5];  // [2]
    float* out        = (float*)d_out;          // [B, 2]
    float* fused      = (float*)d_ws;           // [B, 4] scratch

    const int B = in_sizes[1] / 2;

    // Kernel 1: one wave32 per batch row; 8 waves per 256-thread block
    const int blocks1 = (B + 7) / 8;
    ms_kernel<<<blocks1, 256, 0, stream>>>(von, sinv, fused, B);

    // Kernel 2: one wave32 per 16-row tile
    const int ntiles  = B / 16;
    const int blocks2 = (ntiles + 7) / 8;
    mlp_wmma_kernel<<<blocks2, 256, 0, stream>>>(fused, W1, b1, W2, b2, out, ntiles);
}